// PixelMixer_64484638982763
// MI455X (gfx1250) — compile-verified
//
#include <hip/hip_runtime.h>

// PixelMixer: depthwise circular-shift permutation, [8,512,512,60] fp32.
// Pure data movement: ~503 MB in + ~503 MB out -> HBM-bound (~43 us floor
// at 23.3 TB/s). Strategy: async global->LDS staging (b128, ASYNCcnt) of
// three contiguous row-chunks, LDS gather for the 5-way channel select,
// non-temporal b128 stores.

#define BATCH 8
#define HH    512
#define WW    512
#define CC    60
#define TW    64          // w-positions per block
#define VPW   (CC / 4)    // 15 float4 per w position
#define NTHREADS 256

typedef float v4f __attribute__((ext_vector_type(4)));

// Match the builtin's parameter types exactly: int4 pointers in AS1 / AS3.
typedef int vsi4 __attribute__((vector_size(16)));
typedef __attribute__((address_space(1))) vsi4 global_vsi4;
typedef __attribute__((address_space(3))) vsi4 lds_vsi4;

#if __has_builtin(__builtin_amdgcn_global_load_async_to_lds_b128) && \
    __has_builtin(__builtin_amdgcn_s_wait_asynccnt)
#define USE_ASYNC_LDS 1
#endif

__device__ __forceinline__ void stage16(float* lds_dst, const float* g_src) {
#ifdef USE_ASYNC_LDS
  __builtin_amdgcn_global_load_async_to_lds_b128(
      (global_vsi4*)(g_src),
      (lds_vsi4*)(lds_dst),
      /*imm offset*/ 0, /*cpol*/ 0);
#else
  *(v4f*)lds_dst = *(const v4f*)g_src;
#endif
}

__global__ __launch_bounds__(NTHREADS)
void pixel_mixer_kernel(const float* __restrict__ x, float* __restrict__ out) {
  // Center strip holds w0-1 .. w0+TW (TW+2 positions); top = row h-1, bot = row h+1.
  __shared__ float sC[(TW + 2) * CC];
  __shared__ float sT[TW * CC];
  __shared__ float sB[TW * CC];

  const int tile = blockIdx.x;          // 0 .. WW/TW-1
  const int h    = blockIdx.y;          // 0 .. 511
  const int b    = blockIdx.z;          // 0 .. 7
  const int w0   = tile * TW;
  const int tid  = threadIdx.x;

  const size_t rowPitch = (size_t)WW * CC;                 // floats per (b,h) row
  const size_t imgBase  = (size_t)b * HH * rowPitch;

  const int hT = (h == 0) ? (HH - 1) : (h - 1);
  const int hB = (h == HH - 1) ? 0 : (h + 1);

  const float* rowC = x + imgBase + (size_t)h  * rowPitch;                    // indexed by absolute w
  const float* rowT = x + imgBase + (size_t)hT * rowPitch + (size_t)w0 * CC;  // contiguous chunk
  const float* rowB = x + imgBase + (size_t)hB * rowPitch + (size_t)w0 * CC;  // contiguous chunk

  // ---- Stage center strip: (TW+2)*VPW float4s, w wraps circularly ----
  for (int i = tid; i < (TW + 2) * VPW; i += NTHREADS) {
    const int wo = i / VPW;             // 0 .. TW+1  (maps to src w = w0-1+wo)
    const int q  = i % VPW;
    int sw = w0 - 1 + wo;
    sw = (sw < 0) ? (WW - 1) : ((sw >= WW) ? (sw - WW) : sw);
    stage16(&sC[i * 4], rowC + (size_t)sw * CC + 4 * q);
  }
  // ---- Stage top / bottom strips: fully contiguous ----
  for (int i = tid; i < TW * VPW; i += NTHREADS) {
    stage16(&sT[i * 4], rowT + 4 * (size_t)i);
    stage16(&sB[i * 4], rowB + 4 * (size_t)i);
  }

#ifdef USE_ASYNC_LDS
  __builtin_amdgcn_s_wait_asynccnt(0);  // this wave's async LDS writes landed
#endif
  __syncthreads();                      // all waves' staging visible

  // ---- Gather + store: TW*VPW output float4s ----
  float* outRow = out + imgBase + (size_t)h * rowPitch + (size_t)w0 * CC;
  for (int j = tid; j < TW * VPW; j += NTHREADS) {
    const int wl = j / VPW;             // local w in [0, TW)
    const int q  = j % VPW;
    const int c0 = 4 * q;
    v4f r;
#pragma unroll
    for (int u = 0; u < 4; ++u) {
      const int c = c0 + u;
      const int k = c % 5;
      float v;
      switch (k) {
        case 0:  v = sC[(wl + 2) * CC + c]; break;  // roll -1 along W: x[w+1]
        case 1:  v = sC[(wl    ) * CC + c]; break;  // roll +1 along W: x[w-1]
        case 2:  v = sB[(wl    ) * CC + c]; break;  // roll -1 along H: x[h+1]
        case 3:  v = sT[(wl    ) * CC + c]; break;  // roll +1 along H: x[h-1]
        default: v = sC[(wl + 1) * CC + c]; break;  // identity
      }
      r[u] = v;
    }
    __builtin_nontemporal_store(r, (v4f*)(outRow + (size_t)wl * CC + c0));
  }
}

extern "C" void kernel_launch(void* const* d_in, const int* in_sizes, int n_in,
                              void* d_out, int out_size, void* d_ws, size_t ws_size,
                              hipStream_t stream) {
  const float* x = (const float*)d_in[0];
  float* out     = (float*)d_out;
  (void)in_sizes; (void)n_in; (void)out_size; (void)d_ws; (void)ws_size;

  dim3 grid(WW / TW, HH, BATCH);        // 8 x 512 x 8 = 32768 blocks
  pixel_mixer_kernel<<<grid, NTHREADS, 0, stream>>>(x, out);
}